// GaussAdapt_21586505630197
// MI455X (gfx1250) — compile-verified
//
#include <hip/hip_runtime.h>
#include <hip/hip_fp16.h>

typedef __attribute__((ext_vector_type(2))) float v2f;
typedef __attribute__((ext_vector_type(8))) float v8f;

static constexpr int BN = 4096;    // batch
static constexpr int KN = 1000;    // classes
static constexpr int DN = 1024;    // feature dim
static constexpr int SN = 8;       // memory slots per class
static constexpr int RN = KN * SN; // 8000 rows of centered matrix C
static constexpr int NS_ITERS = 14;

// scalar-buffer slots
#define S_N        0   // total stored samples (float)
#define S_TRACE    1   // trace(M)
#define S_UPDATED  2   // any eviction happened
#define S_SIGANY   3   // any class count > 2
#define S_ROWMAX   4   // ||A||_inf (atomicMax on uint bits)
#define S_ALPHAM   5   // 1 / max(n-1, 1)
#define S_SCALEINV 7   // d if ridge branch else 1
#define S_X0       8   // 1 / ||A||_inf
#define S_RIDGE    9   // 1 if n <= 4d

// ---------------------------------------------------------------- init
__global__ void k_init(float* sbuf, float* colsum) {
    for (int i = threadIdx.x; i < 64; i += blockDim.x) sbuf[i] = 0.f;
    for (int i = threadIdx.x; i < DN; i += blockDim.x) colsum[i] = 0.f;
}

// zero the padding rows KN..DN-1 of musN (so padded GEMM rows are clean)
__global__ void k_zero_pad(float* __restrict__ musN) {
    long idx = (long)blockIdx.x * blockDim.x + threadIdx.x;
    long total = (long)(DN - KN) * DN;
    if (idx < total) musN[(long)KN * DN + idx] = 0.f;
}

// ------------------------------------------- stage 1: eviction scan (per label)
__global__ void k_scan(const int* __restrict__ labels, const float* __restrict__ zent,
                       const __half* __restrict__ ment_in, const unsigned char* __restrict__ st_in,
                       float* sbuf, int* __restrict__ cnt,
                       int* __restrict__ st_out, int* __restrict__ src_out) {
    int k = blockIdx.x * blockDim.x + threadIdx.x;
    if (k >= KN) return;
    float ent[SN]; int st[SN]; int src[SN];
    #pragma unroll
    for (int s = 0; s < SN; ++s) {
        ent[s] = __half2float(ment_in[k * SN + s]);
        st[s]  = st_in[k * SN + s] ? 1 : 0;
        src[s] = -1;
    }
    int anydo = 0;
    for (int i = 0; i < BN; ++i) {
        if (labels[i] != k) continue;
        float e = zent[i];
        int idx = 0; float mv = ent[0];
        #pragma unroll
        for (int s = 1; s < SN; ++s) if (ent[s] > mv) { mv = ent[s]; idx = s; }
        if (e < mv) {                                   // do = any(e < row)
            ent[idx] = __half2float(__float2half(e));   // stored as fp16
            st[idx] = 1; src[idx] = i; anydo = 1;
        }
    }
    int c = 0;
    #pragma unroll
    for (int s = 0; s < SN; ++s) {
        st_out[k * SN + s] = st[s];
        src_out[k * SN + s] = src[s];
        c += st[s];
    }
    cnt[k] = c;
    atomicAdd(&sbuf[S_N], (float)c);
    if (anydo) sbuf[S_UPDATED] = 1.0f;   // racy same-value store: deterministic
    if (c > 2) sbuf[S_SIGANY] = 1.0f;
}

// slot value as f32 (feature rounded through fp16 to mimic memory dtype)
__device__ __forceinline__ float slot_val(const float* feats, const __half* mem_in,
                                          const int* src, int r, int d) {
    int sr = src[r];
    if (sr >= 0) return __half2float(__float2half(feats[(long)sr * DN + d]));
    return __half2float(mem_in[(long)r * DN + d]);
}

// ------------------------------------------- stage 2: class means + column sums
__global__ void k_class_stats(const float* __restrict__ feats, const __half* __restrict__ mem_in,
                              const float* __restrict__ mus_in,
                              const int* __restrict__ st, const int* __restrict__ src,
                              const int* __restrict__ cnt, const float* __restrict__ sbuf,
                              float* __restrict__ mus_new, float* __restrict__ colsum) {
    int k = blockIdx.x;
    int c = cnt[k];
    bool useMean = (c >= 2) && (sbuf[S_UPDATED] > 0.5f);
    for (int d = threadIdx.x; d < DN; d += blockDim.x) {
        float sum = 0.f;
        #pragma unroll
        for (int s = 0; s < SN; ++s) {
            int r = k * SN + s;
            if (st[r]) sum += slot_val(feats, mem_in, src, r, d);
        }
        float mu = useMean ? (sum / (float)SN) : mus_in[(long)k * DN + d];
        mus_new[(long)k * DN + d] = mu;
        atomicAdd(&colsum[d], sum - (float)c * mu);   // sum_s st*(val - mu)
    }
}

__global__ void k_finalize1(float* sbuf, const float* colsum, float* cmean) {
    float n = sbuf[S_N];
    if (threadIdx.x == 0) {
        sbuf[S_ALPHAM] = 1.0f / fmaxf(n - 1.0f, 1.0f);
        float ridge = (n <= 4.0f * (float)DN) ? 1.0f : 0.0f;
        sbuf[S_RIDGE] = ridge;
        sbuf[S_SCALEINV] = (ridge > 0.5f) ? (float)DN : 1.0f;
    }
    float invn = 1.0f / fmaxf(n, 1.0f);
    for (int d = threadIdx.x; d < DN; d += blockDim.x) cmean[d] = colsum[d] * invn;
}

// ------------------------------------------- stage 3: build C^T [DN x RN]
__global__ void k_build_ct(const float* __restrict__ feats, const __half* __restrict__ mem_in,
                           const float* __restrict__ mus_new, const float* __restrict__ cmean,
                           const int* __restrict__ st, const int* __restrict__ src,
                           float* __restrict__ Ct) {
    int d = blockIdx.x;
    float cm = cmean[d];
    for (int r = threadIdx.x; r < RN; r += blockDim.x) {
        float v = 0.f;
        if (st[r]) {
            int k = r >> 3;
            v = slot_val(feats, mem_in, src, r, d) - mus_new[(long)k * DN + d] - cm;
        }
        Ct[(long)d * RN + r] = v;
    }
}

// ------------------------------------------- generic f32 WMMA GEMM (B transposed)
// D = alpha * (A @ B^T) + beta*Cin [+ rowAdd[m] + colAdd[n]]
// A: [M,Kd] row-major lda; B stored row-major ldb, used as element(k,n) = B[n*ldb+k].
// ALL loads must be in-bounds (caller pads buffers). Stores guarded by (r<M && c<N).
// Kd must be a multiple of 4 and >= 8. Wave tile: 32x32 (2x2 fragments).
// Inner loop is software-pipelined: next step's fragments preload into fresh
// registers while current step's 4 WMMAs issue (keeps >=4 loads in flight).
__global__ void k_gemm(const float* __restrict__ A, long lda,
                       const float* __restrict__ B, long ldb,
                       const float* __restrict__ Cin, long ldc,
                       float* __restrict__ Dout, long ldd,
                       int M, int N, int Kd,
                       float alphaImm, const float* __restrict__ alphaPtr,
                       float beta,
                       const float* __restrict__ rowAdd, const float* __restrict__ colAdd) {
    int wave = threadIdx.x >> 5;
    int lane = threadIdx.x & 31;
    int nTn = (N + 31) >> 5;
    int nTm = (M + 31) >> 5;
    int tile = blockIdx.x * (blockDim.x >> 5) + wave;
    if (tile >= nTm * nTn) return;                 // wave-uniform: EXEC stays full
    int tm = (tile / nTn) << 5;
    int tn = (tile % nTn) << 5;
    int ml = lane & 15;        // A row lane / B col lane / D col lane
    int kh = lane >> 4;        // K-half select (2 consecutive k per lane)
    const float* A0 = A + (long)(tm + ml) * lda + 2 * kh;
    const float* A1 = A0 + (long)16 * lda;
    const float* B0 = B + (long)(tn + ml) * ldb + 2 * kh;
    const float* B1 = B0 + (long)16 * ldb;
    v8f acc00 = {0.f,0.f,0.f,0.f,0.f,0.f,0.f,0.f};
    v8f acc01 = acc00, acc10 = acc00, acc11 = acc00;
    // prologue: fragments for k-step 0
    v2f a0 = *reinterpret_cast<const v2f*>(A0);
    v2f a1 = *reinterpret_cast<const v2f*>(A1);
    v2f b0 = *reinterpret_cast<const v2f*>(B0);
    v2f b1 = *reinterpret_cast<const v2f*>(B1);
    // steady state: preload (kk+4) then consume (kk); preload always in-bounds
    #pragma unroll 4
    for (int kk = 0; kk < Kd - 4; kk += 4) {
        long on = kk + 4;
        if ((kk & 63) == 0) {                                   // near-cache prefetch ahead
            long pf = (kk + 128 < Kd) ? (long)(kk + 128) : 0l;
            __builtin_prefetch(A0 + pf, 0, 3);
            __builtin_prefetch(B0 + pf, 0, 3);
        }
        v2f na0 = *reinterpret_cast<const v2f*>(A0 + on);
        v2f na1 = *reinterpret_cast<const v2f*>(A1 + on);
        v2f nb0 = *reinterpret_cast<const v2f*>(B0 + on);
        v2f nb1 = *reinterpret_cast<const v2f*>(B1 + on);
        acc00 = __builtin_amdgcn_wmma_f32_16x16x4_f32(false, a0, false, b0, (short)0, acc00, false, false);
        acc01 = __builtin_amdgcn_wmma_f32_16x16x4_f32(false, a0, false, b1, (short)0, acc01, false, false);
        acc10 = __builtin_amdgcn_wmma_f32_16x16x4_f32(false, a1, false, b0, (short)0, acc10, false, false);
        acc11 = __builtin_amdgcn_wmma_f32_16x16x4_f32(false, a1, false, b1, (short)0, acc11, false, false);
        a0 = na0; a1 = na1; b0 = nb0; b1 = nb1;
    }
    // epilogue k-step (last 4)
    acc00 = __builtin_amdgcn_wmma_f32_16x16x4_f32(false, a0, false, b0, (short)0, acc00, false, false);
    acc01 = __builtin_amdgcn_wmma_f32_16x16x4_f32(false, a0, false, b1, (short)0, acc01, false, false);
    acc10 = __builtin_amdgcn_wmma_f32_16x16x4_f32(false, a1, false, b0, (short)0, acc10, false, false);
    acc11 = __builtin_amdgcn_wmma_f32_16x16x4_f32(false, a1, false, b1, (short)0, acc11, false, false);

    float alpha = alphaImm * (alphaPtr ? *alphaPtr : 1.0f);
    int mo = kh << 3;
    #pragma unroll
    for (int t = 0; t < 4; ++t) {
        const v8f& acc = (t == 0) ? acc00 : (t == 1) ? acc01 : (t == 2) ? acc10 : acc11;
        int r0 = tm + ((t >> 1) << 4) + mo;
        int c  = tn + ((t & 1) << 4) + ml;
        #pragma unroll
        for (int j = 0; j < 8; ++j) {
            int r = r0 + j;
            if (r < M && c < N) {
                float v = alpha * acc[j];
                if (Cin)    v += beta * Cin[(long)r * ldc + c];
                if (rowAdd) v += rowAdd[r];
                if (colAdd) v += colAdd[c];
                Dout[(long)r * ldd + c] = v;
            }
        }
    }
}

// ------------------------------------------- trace(M)
__global__ void k_trace(const float* __restrict__ Mm, float* sbuf) {
    __shared__ float red[256];
    float s = 0.f;
    for (int d = threadIdx.x; d < DN; d += blockDim.x) s += Mm[(long)d * DN + d];
    red[threadIdx.x] = s; __syncthreads();
    for (int off = 128; off > 0; off >>= 1) {
        if ((int)threadIdx.x < off) red[threadIdx.x] += red[threadIdx.x + off];
        __syncthreads();
    }
    if (threadIdx.x == 0) sbuf[S_TRACE] = red[0];
}

// ------------------------------------------- A = (n-1)M + tr*I (ridge) or M; ||A||_inf
__global__ void k_build_A(const float* __restrict__ Mm, float* sbuf, float* __restrict__ Aout) {
    int i = blockIdx.x;
    float n = sbuf[S_N], tr = sbuf[S_TRACE];
    bool ridge = sbuf[S_RIDGE] > 0.5f;
    __shared__ float red[256];
    float rs = 0.f;
    for (int j = threadIdx.x; j < DN; j += blockDim.x) {
        float v = Mm[(long)i * DN + j];
        if (ridge) v = (n - 1.0f) * v + ((i == j) ? tr : 0.f);
        Aout[(long)i * DN + j] = v;
        rs += fabsf(v);
    }
    red[threadIdx.x] = rs; __syncthreads();
    for (int off = 128; off > 0; off >>= 1) {
        if ((int)threadIdx.x < off) red[threadIdx.x] += red[threadIdx.x + off];
        __syncthreads();
    }
    if (threadIdx.x == 0)
        atomicMax(reinterpret_cast<unsigned int*>(&sbuf[S_ROWMAX]), __float_as_uint(red[0]));
}

__global__ void k_finalize3(float* sbuf) {
    float rm = sbuf[S_ROWMAX];
    sbuf[S_X0] = 1.0f / fmaxf(rm, 1e-30f);
}

__global__ void k_setX0(const float* __restrict__ sbuf, float* __restrict__ X) {
    int i = blockIdx.x;
    float x0 = sbuf[S_X0];
    for (int j = threadIdx.x; j < DN; j += blockDim.x)
        X[(long)i * DN + j] = (i == j) ? x0 : 0.f;
}

// inv_used = sig_gate ? scale*X : inv_Sig_in
__global__ void k_select_inv(const float* __restrict__ X, const float* __restrict__ inv_in,
                             const float* __restrict__ sbuf, float* __restrict__ out) {
    int i = blockIdx.x;
    bool gate = (sbuf[S_SIGANY] > 0.5f) && (sbuf[S_UPDATED] > 0.5f);
    float sc = sbuf[S_SCALEINV];
    for (int j = threadIdx.x; j < DN; j += blockDim.x)
        out[(long)i * DN + j] = gate ? sc * X[(long)i * DN + j] : inv_in[(long)i * DN + j];
}

// out[row] = scale * dot(A[row,:], B[row,:])
__global__ void k_rowdot(const float* __restrict__ A, const float* __restrict__ B,
                         float* __restrict__ out, int D, float scale) {
    __shared__ float red[256];
    long row = blockIdx.x;
    float s = 0.f;
    for (int j = threadIdx.x; j < D; j += blockDim.x)
        s += A[row * D + j] * B[row * D + j];
    red[threadIdx.x] = s; __syncthreads();
    for (int off = 128; off > 0; off >>= 1) {
        if ((int)threadIdx.x < off) red[threadIdx.x] += red[threadIdx.x + off];
        __syncthreads();
    }
    if (threadIdx.x == 0) out[row] = scale * red[0];
}

// ---------------------------------------------------------------- host
static inline int gemm_blocks(int M, int N) {
    int tiles = ((M + 31) / 32) * ((N + 31) / 32);
    return (tiles + 7) / 8;   // 8 waves (32x32 tiles) per 256-thread block
}

extern "C" void kernel_launch(void* const* d_in, const int* in_sizes, int n_in,
                              void* d_out, int out_size, void* d_ws, size_t ws_size,
                              hipStream_t stream) {
    const float*         feats   = (const float*)d_in[0];          // [B,D]
    const float*         zent    = (const float*)d_in[2];          // [B]
    const int*           labels  = (const int*)d_in[3];            // [B]
    const __half*        mem_in  = (const __half*)d_in[4];         // [K,S,D] f16
    const __half*        ment_in = (const __half*)d_in[5];         // [K,S] f16
    const unsigned char* st_in   = (const unsigned char*)d_in[6];  // [K,S] bool
    const float*         mus_in  = (const float*)d_in[8];          // [K,D]
    const float*         inv_in  = (const float*)d_in[10];         // [D,D]
    float*               out     = (float*)d_out;                  // [B,K]

    char* w = (char*)d_ws;
    size_t off = 0;
    auto alloc = [&](size_t bytes) -> void* {
        void* p = w + off;
        off = (off + bytes + 255) & ~((size_t)255);
        return p;
    };
    float* sbuf   = (float*)alloc(64 * sizeof(float));
    float* colsum = (float*)alloc(DN * sizeof(float));
    float* cmean  = (float*)alloc(DN * sizeof(float));
    int*   cnt    = (int*)alloc(KN * sizeof(int));
    int*   stbuf  = (int*)alloc(RN * sizeof(int));
    int*   srcbuf = (int*)alloc(RN * sizeof(int));
    float* musN   = (float*)alloc((size_t)DN * DN * sizeof(float)); // padded to 1024 rows
    float* Mm     = (float*)alloc((size_t)DN * DN * sizeof(float));
    float* Ans    = (float*)alloc((size_t)DN * DN * sizeof(float));
    float* Xa     = (float*)alloc((size_t)DN * DN * sizeof(float));
    float* Xb     = (float*)alloc((size_t)DN * DN * sizeof(float));
    float* Tm     = (float*)alloc((size_t)DN * DN * sizeof(float));
    float* invU   = (float*)alloc((size_t)DN * DN * sizeof(float));
    float* Wt     = (float*)alloc((size_t)DN * DN * sizeof(float)); // padded to 1024 rows
    float* bvec   = (float*)alloc(KN * sizeof(float));
    float* Qvec   = (float*)alloc(BN * sizeof(float));
    float* Ct     = (float*)alloc((size_t)DN * RN * sizeof(float)); // reused as Y later
    float* Y      = Ct;   // alias: Ct dead after M is computed

    k_init<<<1, 256, 0, stream>>>(sbuf, colsum);
    k_zero_pad<<<((DN - KN) * DN + 255) / 256, 256, 0, stream>>>(musN);

    // stage 1: eviction scan (1 thread per class)
    k_scan<<<(KN + 255) / 256, 256, 0, stream>>>(labels, zent, ment_in, st_in,
                                                 sbuf, cnt, stbuf, srcbuf);
    // stage 2: class means + global column sums -> cmean
    k_class_stats<<<KN, 256, 0, stream>>>(feats, mem_in, mus_in, stbuf, srcbuf,
                                          cnt, sbuf, musN, colsum);
    k_finalize1<<<1, 256, 0, stream>>>(sbuf, colsum, cmean);

    // stage 3: Ct [DN x RN], then M = (1/(n-1)) * Ct @ Ct^T   (WMMA)
    k_build_ct<<<DN, 256, 0, stream>>>(feats, mem_in, musN, cmean, stbuf, srcbuf, Ct);
    k_gemm<<<gemm_blocks(DN, DN), 256, 0, stream>>>(
        Ct, (long)RN, Ct, (long)RN, nullptr, 0, Mm, (long)DN,
        DN, DN, RN, 1.0f, &sbuf[S_ALPHAM], 0.f, nullptr, nullptr);

    // stage 4: trace, ridge matrix A, ||A||_inf, X0 = I/||A||_inf
    k_trace<<<1, 256, 0, stream>>>(Mm, sbuf);
    k_build_A<<<DN, 256, 0, stream>>>(Mm, sbuf, Ans);
    k_finalize3<<<1, 1, 0, stream>>>(sbuf);
    k_setX0<<<DN, 256, 0, stream>>>(sbuf, Xa);

    // stage 5: Newton-Schulz X <- 2X - X A X, exploiting symmetry of A and X:
    //   T2 = X @ A   (= X @ A^T since A symmetric)
    //   Xn = -T2 @ X + 2X  (= -T2 @ X^T since X symmetric)
    float* X = Xa; float* Xn = Xb;
    for (int it = 0; it < NS_ITERS; ++it) {
        k_gemm<<<gemm_blocks(DN, DN), 256, 0, stream>>>(
            X, (long)DN, Ans, (long)DN, nullptr, 0, Tm, (long)DN,
            DN, DN, DN, 1.0f, nullptr, 0.f, nullptr, nullptr);
        k_gemm<<<gemm_blocks(DN, DN), 256, 0, stream>>>(
            Tm, (long)DN, X, (long)DN, X, (long)DN, Xn, (long)DN,
            DN, DN, DN, -1.0f, nullptr, 2.0f, nullptr, nullptr);
        float* tmp = X; X = Xn; Xn = tmp;
    }

    // stage 6: select inverse (d * ridge_inv, or mp_inv, or original inv_Sig)
    k_select_inv<<<DN, 256, 0, stream>>>(X, inv_in, sbuf, invU);

    // stage 7: Wt = mus_new @ invU^T (invU symmetric); padded rows give Wt pad rows
    k_gemm<<<gemm_blocks(DN, DN), 256, 0, stream>>>(
        musN, (long)DN, invU, (long)DN, nullptr, 0, Wt, (long)DN,
        DN, DN, DN, 1.0f, nullptr, 0.f, nullptr, nullptr);
    k_rowdot<<<KN, 256, 0, stream>>>(musN, Wt, bvec, DN, -0.5f);

    // stage 8: Y = features @ invU^T ; Q[n] = -0.5 f[n].Y[n]
    k_gemm<<<gemm_blocks(BN, DN), 256, 0, stream>>>(
        feats, (long)DN, invU, (long)DN, nullptr, 0, Y, (long)DN,
        BN, DN, DN, 1.0f, nullptr, 0.f, nullptr, nullptr);
    k_rowdot<<<BN, 256, 0, stream>>>(feats, Y, Qvec, DN, -0.5f);

    // stage 9: logits = features @ Wt^T + b[col] + Q[row]  -> d_out
    k_gemm<<<gemm_blocks(BN, KN), 256, 0, stream>>>(
        feats, (long)DN, Wt, (long)DN, nullptr, 0, out, (long)KN,
        BN, KN, DN, 1.0f, nullptr, 0.f, Qvec, bvec);
}